// DenseCaptioningLoss_38749194944940
// MI455X (gfx1250) — compile-verified
//
#include <hip/hip_runtime.h>
#include <hip/hip_bf16.h>

// Dense-captioning loss for MI455X (gfx1250, wave32).
// Bandwidth-bound: ~162 MB of fp32 logits streamed once (b128 loads),
// per-row logsumexp in registers, deterministic final reduction via
// V_WMMA_F32_16X16X4_F32 (B = ones -> row sums).

#define L2E 1.4426950408889634f
#define LN2 0.6931471805599453f

typedef float v2f __attribute__((ext_vector_type(2)));
typedef float v8f __attribute__((ext_vector_type(8)));

__device__ __forceinline__ float waveRedSum(float v) {
#pragma unroll
  for (int o = 16; o > 0; o >>= 1) v += __shfl_xor(v, o, 32);
  return v;
}
__device__ __forceinline__ float waveRedMax(float v) {
#pragma unroll
  for (int o = 16; o > 0; o >>= 1) v = fmaxf(v, __shfl_xor(v, o, 32));
  return v;
}

// One block per (item, token) row. Writes -w * logsoftmax[row, tgt] (or 0 if
// masked) to partials[row]. V = vocab, NV4 = V/4, KITER = ceil(NV4/256).
// CDIM > 0 => caption mode (lens indexed by b*CDIM+c, item mask from counts);
// CDIM == 0 => program mode (lens indexed by b).
template <int V, int NV4, int KITER, int LDIM, int CDIM>
__global__ __launch_bounds__(256) void sentlen_partial_kernel(
    const float* __restrict__ logits, const int* __restrict__ targets,
    const int* __restrict__ lens, const int* __restrict__ counts,
    float* __restrict__ partials) {
  const int r = blockIdx.x;
  const int tid = threadIdx.x;
  int len;
  bool valid;
  if constexpr (CDIM > 0) {
    const int l = r % LDIM;
    const int bc = r / LDIM;
    const int c = bc % CDIM;
    const int b = bc / CDIM;
    len = lens[bc];
    valid = (l < len) && (c < counts[b]);
  } else {
    const int l = r % LDIM;
    const int b = r / LDIM;
    len = lens[b];
    valid = (l < len);
  }
  if (!valid) {  // uniform per block
    if (tid == 0) partials[r] = 0.0f;
    return;
  }

  const float4* __restrict__ row4 =
      (const float4*)(logits + (size_t)r * (size_t)V);
  float4 vals[KITER];
  float m = -__builtin_inff();
#pragma unroll
  for (int k = 0; k < KITER; ++k) {
    const int i = tid + k * 256;
    if (i < NV4) {
      const float4 v = row4[i];  // global_load_b128
      vals[k] = v;
      m = fmaxf(m, fmaxf(fmaxf(v.x, v.y), fmaxf(v.z, v.w)));
    }
  }

  __shared__ float red[8];
  __shared__ float Msh, Ssh, Xsh;
  const int lane = tid & 31;
  const int wv = tid >> 5;

  // block max
  float t = waveRedMax(m);
  if (lane == 0) red[wv] = t;
  __syncthreads();
  if (wv == 0) {
    float q = (lane < 8) ? red[lane] : -__builtin_inff();
    q = waveRedMax(q);
    if (lane == 0) Msh = q;
  }
  __syncthreads();
  const float M = Msh;

  // block sum of exp2((x - M) * log2(e))
  float s = 0.0f;
#pragma unroll
  for (int k = 0; k < KITER; ++k) {
    const int i = tid + k * 256;
    if (i < NV4) {
      const float4 v = vals[k];
      s += __builtin_amdgcn_exp2f((v.x - M) * L2E);
      s += __builtin_amdgcn_exp2f((v.y - M) * L2E);
      s += __builtin_amdgcn_exp2f((v.z - M) * L2E);
      s += __builtin_amdgcn_exp2f((v.w - M) * L2E);
    }
  }
  __syncthreads();  // red[] reuse
  t = waveRedSum(s);
  if (lane == 0) red[wv] = t;
  __syncthreads();
  if (wv == 0) {
    float q = (lane < 8) ? red[lane] : 0.0f;
    q = waveRedSum(q);
    if (lane == 0) Ssh = q;
  }

  // target logit lives in exactly one thread's registers
  const int tgt = targets[r];
  const int i4 = tgt >> 2;
  if ((i4 & 255) == tid) {
    const int ksel = i4 >> 8;
    float4 v = vals[0];
#pragma unroll
    for (int k = 1; k < KITER; ++k)
      if (k == ksel) v = vals[k];
    const int cpn = tgt & 3;
    Xsh = (cpn == 0) ? v.x : (cpn == 1) ? v.y : (cpn == 2) ? v.z : v.w;
  }
  __syncthreads();

  if (tid == 0) {
    const float lse = M + LN2 * __builtin_amdgcn_logf(Ssh);
    const float w =
        __builtin_amdgcn_exp2f(-0.7f * __builtin_amdgcn_logf((float)len));
    partials[r] = -w * (Xsh - lse);
  }
}

// Deterministic sum of n (multiple of 64) floats with V_WMMA_F32_16X16X4_F32:
// D = A(16x4) * ones(4x16) + C  => every D element accumulates its A-row sum;
// summing one D column over all 16 rows yields the grand total (broadcast to
// every lane since all columns are identical).
__device__ __forceinline__ float wmma_sum64(const float* __restrict__ p,
                                            int n, int lane) {
  const v2f bones = {1.0f, 1.0f};
  v8f acc = {0.f, 0.f, 0.f, 0.f, 0.f, 0.f, 0.f, 0.f};
  for (int base = 0; base < n; base += 64) {
    v2f a;
    a.x = p[base + lane];
    a.y = p[base + 32 + lane];
    acc = __builtin_amdgcn_wmma_f32_16x16x4_f32(
        /*neg_a=*/false, a, /*neg_b=*/false, bones,
        /*c_mod=*/(short)0, acc, /*reuse_a=*/false, /*reuse_b=*/false);
  }
  float rs = acc[0] + acc[1] + acc[2] + acc[3] + acc[4] + acc[5] + acc[6] +
             acc[7];              // lanes 0-15: rows 0-7; lanes 16-31: rows 8-15
  rs += __shfl_xor(rs, 16, 32);   // all lanes now hold the total
  return rs;
}

__global__ __launch_bounds__(32) void finalize_kernel(
    const float* __restrict__ ws, const int* __restrict__ counts,
    const float* __restrict__ pred_iv, const float* __restrict__ gt_iv,
    const float* __restrict__ scores, float* __restrict__ out) {
  const int lane = threadIdx.x;  // single wave32, EXEC all ones for WMMA

  const float capS = wmma_sum64(ws, 3840, lane);         // 16*8*30 partials
  const float progS = wmma_sum64(ws + 3840, 1024, lane); // 16*64 partials

  // n_items for captions = sum_b count_b (count in [1,8] = C)
  float nc = (lane < 16) ? (float)counts[lane] : 0.0f;
  nc = waveRedSum(nc);

  // IoU over 128 intervals, 4 per lane
  float is = 0.0f;
#pragma unroll
  for (int k = 0; k < 4; ++k) {
    const int i = lane * 4 + k;
    const float p0 = pred_iv[2 * i], p1 = pred_iv[2 * i + 1];
    const float g0 = gt_iv[2 * i], g1 = gt_iv[2 * i + 1];
    const float inter = fmaxf(fminf(p1, g1) - fmaxf(p0, g0), 0.0f);
    const float uni = fmaxf(p1, g1) - fminf(p0, g0);
    is += inter / uni;
  }
  is = waveRedSum(is);

  if (lane == 0) {
    const float cap_loss = capS / nc;
    const float prog_loss = progS * (1.0f / 16.0f);
    const float iou_loss = 1.0f - is * (1.0f / 128.0f);
    const float loss =
        scores[0] * cap_loss + scores[1] * prog_loss + scores[2] * iou_loss;
    out[0] = loss;
    out[1] = cap_loss;
    out[2] = prog_loss;
    out[3] = iou_loss;
  }
}

extern "C" void kernel_launch(void* const* d_in, const int* in_sizes, int n_in,
                              void* d_out, int out_size, void* d_ws,
                              size_t ws_size, hipStream_t stream) {
  const int* gt_captions = (const int*)d_in[0];     // [16,8,30]
  const int* gt_cap_lens = (const int*)d_in[1];     // [16,8]
  const float* pred_captions = (const float*)d_in[2];  // [16,8,30,10000]
  const int* gt_program = (const int*)d_in[3];      // [16,64]
  const int* gt_prog_len = (const int*)d_in[4];     // [16]
  const float* pred_program = (const float*)d_in[5];   // [16,64,2000]
  const float* gt_intervals = (const float*)d_in[6];   // [128,2,1]
  const float* pred_intervals = (const float*)d_in[7]; // [128,2,1]
  const int* gt_caps_count = (const int*)d_in[8];   // [16]
  const float* scores = (const float*)d_in[9];      // [3]
  float* out = (float*)d_out;
  float* ws = (float*)d_ws;  // [0,3840): cap partials, [3840,4864): prog

  sentlen_partial_kernel<10000, 2500, 10, 30, 8>
      <<<3840, 256, 0, stream>>>(pred_captions, gt_captions, gt_cap_lens,
                                 gt_caps_count, ws);
  sentlen_partial_kernel<2000, 500, 2, 64, 0>
      <<<1024, 256, 0, stream>>>(pred_program, gt_program, gt_prog_len,
                                 nullptr, ws + 3840);
  finalize_kernel<<<1, 32, 0, stream>>>(ws, gt_caps_count, pred_intervals,
                                        gt_intervals, scores, out);
}